// SimpleGCN_50766513438884
// MI455X (gfx1250) — compile-verified
//
#include <hip/hip_runtime.h>

#define N_NODES   50000
#define N_EDGES   800000
#define N_FEAT    140
#define HIDDEN    128
#define N_CLASSES 24
#define N_GRAPHS  64
#define H1        73
#define H2        41

#define KC_MAX     32
#define LDS_STRIDE 136   // 128 + 8: shifts lane-group-1 rows by 16 banks

typedef __attribute__((ext_vector_type(2))) float v2f;
typedef __attribute__((ext_vector_type(8))) float v8f;

// ---------------- utility ----------------
__global__ void fill_f32(float* __restrict__ p, float v, int n) {
    int i = blockIdx.x * blockDim.x + threadIdx.x;
    if (i < n) p[i] = v;
}

// deg[dst] += 1 for each edge (deg pre-initialized to 1.0 for the self-loop)
__global__ void deg_count(const int* __restrict__ dst, float* __restrict__ deg) {
    int i = blockIdx.x * blockDim.x + threadIdx.x;
    if (i < N_EDGES) atomicAdd(&deg[dst[i]], 1.0f);
}

__global__ void rsqrt_inplace(float* __restrict__ d, int n) {
    int i = blockIdx.x * blockDim.x + threadIdx.x;
    if (i < n) d[i] = rsqrtf(d[i]);
}

// -------- fp32 WMMA GEMM with LDS-staged W and 8-way register blocking -----
// D[M,128] = A[M,K] @ W[K,128].  blockDim = 256 (8 waves).  Each wave owns a
// 16-row strip and all 8 column tiles (8 v8f accumulators -> A reused 8x).
// W is staged KC rows at a time into padded LDS. Requires K % KC == 0,
// KC % 4 == 0, KC <= KC_MAX, M % 16 == 0.
__global__ void gemm_f32_wmma_lds(const float* __restrict__ A,
                                  const float* __restrict__ W,
                                  float* __restrict__ D,
                                  int M, int K, int KC) {
    __shared__ float wl[KC_MAX * LDS_STRIDE];
    const int tid   = threadIdx.x;
    const int wave  = tid >> 5;
    const int lane  = tid & 31;
    const int half  = lane >> 4;       // 0/1 -> K-pair select
    const int lm    = lane & 15;
    const int mtile = blockIdx.x * 8 + wave;   // 16-row strip index
    const bool active = (mtile * 16) < M;      // wave-uniform

    v8f acc[8];
#pragma unroll
    for (int t = 0; t < 8; ++t) acc[t] = (v8f){0.f,0.f,0.f,0.f,0.f,0.f,0.f,0.f};

    const float* arow = A + (size_t)((active ? mtile : 0) * 16 + lm) * K;

    for (int k0 = 0; k0 < K; k0 += KC) {
        // cooperative, coalesced stage of W[k0 .. k0+KC) x 128 into LDS
        const int total = KC * 128;              // multiple of 256
        for (int i = tid; i < total; i += 256) {
            int r = i >> 7, c = i & 127;
            wl[r * LDS_STRIDE + c] = W[(size_t)(k0 + r) * HIDDEN + c];
        }
        __syncthreads();

        if (active) {
            for (int k = 0; k < KC; k += 4) {
                v2f a;
                a.x = arow[k0 + k + half * 2];
                a.y = arow[k0 + k + half * 2 + 1];
                const float* wb = &wl[(k + half * 2) * LDS_STRIDE + lm];
#pragma unroll
                for (int t = 0; t < 8; ++t) {
                    v2f b;
                    b.x = wb[t * 16];
                    b.y = wb[t * 16 + LDS_STRIDE];
                    acc[t] = __builtin_amdgcn_wmma_f32_16x16x4_f32(
                        false, a, false, b, (short)0, acc[t], false, false);
                }
            }
        }
        __syncthreads();
    }

    if (active) {
        // D layout: lane l, vgpr i -> row mtile*16 + 8*half + i, col t*16 + lm
        float* drow = D + (size_t)(mtile * 16 + half * 8) * HIDDEN + lm;
#pragma unroll
        for (int t = 0; t < 8; ++t)
#pragma unroll
            for (int i = 0; i < 8; ++i)
                drow[(size_t)i * HIDDEN + t * 16] = acc[t][i];
    }
}

// ---------------- edge scatter: acc[dst] += xw[src] * dinv[src]*dinv[dst] ---
// One wave per edge; edge metadata forced onto the scalar path, lanes move
// the 128-wide feature row as float4 + 4 global f32 atomics.
__global__ void edge_scatter(const float* __restrict__ xw,
                             const float* __restrict__ dinv,
                             const int* __restrict__ src,
                             const int* __restrict__ dst,
                             float* __restrict__ acc) {
    int e = __builtin_amdgcn_readfirstlane(blockIdx.x * 8 + (threadIdx.x >> 5));
    int lane = threadIdx.x & 31;
    if (e >= N_EDGES) return;
    int s = src[e];
    int d = dst[e];
    float c = dinv[s] * dinv[d];
    const float4* xv = (const float4*)(xw + (size_t)s * HIDDEN);
    float4 v = xv[lane];
    float* out = acc + (size_t)d * HIDDEN + lane * 4;
    atomicAdd(out + 0, v.x * c);
    atomicAdd(out + 1, v.y * c);
    atomicAdd(out + 2, v.z * c);
    atomicAdd(out + 3, v.w * c);
}

// -------- h = relu(acc + xw*dinv^2 + bias); re-zero acc for next conv ------
__global__ void conv_finalize(float* __restrict__ acc,
                              const float* __restrict__ xw,
                              const float* __restrict__ dinv,
                              const float* __restrict__ bias,
                              float* __restrict__ h) {
    int idx = blockIdx.x * blockDim.x + threadIdx.x;
    if (idx >= N_NODES * HIDDEN) return;
    int node = idx >> 7;
    int f    = idx & 127;
    float di = dinv[node];
    float a  = acc[idx];
    acc[idx] = 0.0f;                       // ready for next conv's scatter
    h[idx]   = fmaxf(a + xw[idx] * di * di + bias[f], 0.0f);
}

// ---------------- global mean pool ----------------
__global__ void pool_sum(const float* __restrict__ h,
                         const int* __restrict__ batch,
                         float* __restrict__ gsum) {
    int idx = blockIdx.x * blockDim.x + threadIdx.x;
    if (idx >= N_NODES * HIDDEN) return;
    int node = __builtin_amdgcn_readfirstlane(idx >> 7);  // wave-uniform
    int f    = idx & 127;
    atomicAdd(&gsum[(size_t)batch[node] * HIDDEN + f], h[idx]);
}

__global__ void pool_cnt(const int* __restrict__ batch, float* __restrict__ cnt) {
    int i = blockIdx.x * blockDim.x + threadIdx.x;
    if (i < N_NODES) atomicAdd(&cnt[batch[i]], 1.0f);
}

__global__ void pool_div(const float* __restrict__ gsum,
                         const float* __restrict__ cnt,
                         float* __restrict__ g) {
    int idx = blockIdx.x * blockDim.x + threadIdx.x;
    if (idx < N_GRAPHS * HIDDEN) g[idx] = gsum[idx] / fmaxf(cnt[idx >> 7], 1.0f);
}

// ---------------- small dense + relu: out = relu(in @ W + b) ----------------
__global__ void lin_relu(const float* __restrict__ in, const float* __restrict__ W,
                         const float* __restrict__ b, float* __restrict__ out,
                         int M, int K, int N) {
    int idx = blockIdx.x * blockDim.x + threadIdx.x;
    if (idx >= M * N) return;
    int m = idx / N;
    int n = idx - m * N;
    float s = b[n];
    const float* ir = in + (size_t)m * K;
    for (int k = 0; k < K; ++k) s = fmaf(ir[k], W[(size_t)k * N + n], s);
    out[idx] = fmaxf(s, 0.0f);
}

// ---------------- launcher ----------------
extern "C" void kernel_launch(void* const* d_in, const int* in_sizes, int n_in,
                              void* d_out, int out_size, void* d_ws, size_t ws_size,
                              hipStream_t stream) {
    (void)in_sizes; (void)n_in; (void)out_size; (void)ws_size;

    const float* x     = (const float*)d_in[0];
    const int*   edge  = (const int*)d_in[1];
    const int*   src   = edge;             // edge_index[0]
    const int*   dst   = edge + N_EDGES;   // edge_index[1]
    const int*   batch = (const int*)d_in[2];
    const float* W_in  = (const float*)d_in[3];
    const float* b_in  = (const float*)d_in[4];
    const float* W_mid = (const float*)d_in[5];
    const float* b_mid = (const float*)d_in[6];
    const float* W1    = (const float*)d_in[7];
    const float* b1    = (const float*)d_in[8];
    const float* W2    = (const float*)d_in[9];
    const float* b2    = (const float*)d_in[10];
    const float* W3    = (const float*)d_in[11];
    const float* b3    = (const float*)d_in[12];
    float* out = (float*)d_out;

    // workspace carve-up (floats)
    float* ws     = (float*)d_ws;
    float* bufXW  = ws;                                  // N_NODES*HIDDEN
    float* bufACC = bufXW  + (size_t)N_NODES * HIDDEN;   // N_NODES*HIDDEN
    float* bufH   = bufACC + (size_t)N_NODES * HIDDEN;   // N_NODES*HIDDEN
    float* dinv   = bufH   + (size_t)N_NODES * HIDDEN;   // N_NODES
    float* gsum   = dinv + N_NODES;                      // N_GRAPHS*HIDDEN
    float* cnt    = gsum + N_GRAPHS * HIDDEN;            // N_GRAPHS
    float* g      = cnt  + N_GRAPHS;                     // N_GRAPHS*HIDDEN
    float* g1     = g    + N_GRAPHS * HIDDEN;            // N_GRAPHS*H1
    float* g2     = g1   + N_GRAPHS * H1;                // N_GRAPHS*H2

    const int NH = N_NODES * HIDDEN;
    const int gemm_blocks = (N_NODES / 16 + 7) / 8;      // 3125 waves -> 391 blocks
    const int scat_blocks = (N_EDGES + 7) / 8;           // 8 waves/block, 1 edge/wave

    // deg = 1 (self loop) + in-degree; dinv = rsqrt(deg)
    fill_f32<<<(N_NODES + 255) / 256, 256, 0, stream>>>(dinv, 1.0f, N_NODES);
    deg_count<<<(N_EDGES + 255) / 256, 256, 0, stream>>>(dst, dinv);
    rsqrt_inplace<<<(N_NODES + 255) / 256, 256, 0, stream>>>(dinv, N_NODES);

    // zero the scatter accumulator once; conv_finalize re-zeroes it each conv
    fill_f32<<<(NH + 255) / 256, 256, 0, stream>>>(bufACC, 0.0f, NH);

    // conv1: x @ W_in  (K = 140, KC = 28 -> 5 chunks)
    gemm_f32_wmma_lds<<<gemm_blocks, 256, 0, stream>>>(x, W_in, bufXW, N_NODES, N_FEAT, 28);
    edge_scatter<<<scat_blocks, 256, 0, stream>>>(bufXW, dinv, src, dst, bufACC);
    conv_finalize<<<(NH + 255) / 256, 256, 0, stream>>>(bufACC, bufXW, dinv, b_in, bufH);

    // conv_mid x3 (same weights), K = 128, KC = 32 -> 4 chunks
    for (int i = 0; i < 3; ++i) {
        gemm_f32_wmma_lds<<<gemm_blocks, 256, 0, stream>>>(bufH, W_mid, bufXW, N_NODES, HIDDEN, 32);
        edge_scatter<<<scat_blocks, 256, 0, stream>>>(bufXW, dinv, src, dst, bufACC);
        conv_finalize<<<(NH + 255) / 256, 256, 0, stream>>>(bufACC, bufXW, dinv, b_mid, bufH);
    }

    // global mean pool
    fill_f32<<<(N_GRAPHS * HIDDEN + 255) / 256, 256, 0, stream>>>(gsum, 0.0f, N_GRAPHS * HIDDEN);
    fill_f32<<<1, 64, 0, stream>>>(cnt, 0.0f, N_GRAPHS);
    pool_sum<<<(NH + 255) / 256, 256, 0, stream>>>(bufH, batch, gsum);
    pool_cnt<<<(N_NODES + 255) / 256, 256, 0, stream>>>(batch, cnt);
    pool_div<<<(N_GRAPHS * HIDDEN + 255) / 256, 256, 0, stream>>>(gsum, cnt, g);

    // MLP head
    lin_relu<<<(N_GRAPHS * H1 + 255) / 256, 256, 0, stream>>>(g,  W1, b1, g1,  N_GRAPHS, HIDDEN, H1);
    lin_relu<<<(N_GRAPHS * H2 + 255) / 256, 256, 0, stream>>>(g1, W2, b2, g2,  N_GRAPHS, H1, H2);
    lin_relu<<<(N_GRAPHS * N_CLASSES + 255) / 256, 256, 0, stream>>>(g2, W3, b3, out, N_GRAPHS, H2, N_CLASSES);
}